// CoxPHLossWithTies_3006477107398
// MI455X (gfx1250) — compile-verified
//
#include <hip/hip_runtime.h>
#include <hip/hip_bf16.h>

#define NBINS 2048
#define SCAL_OFF (4 * NBINS)   // 8192: hrE, eCnt, corr

typedef __attribute__((ext_vector_type(2))) float v2f;
typedef __attribute__((ext_vector_type(8))) float v8f;

// ---------------------------------------------------------------------------
// Kernel 0: zero the workspace accumulators (ws is poisoned by the harness).
// ---------------------------------------------------------------------------
__global__ void cox_init(float* __restrict__ ws) {
    int i = blockIdx.x * blockDim.x + threadIdx.x;
    int stride = blockDim.x * gridDim.x;
    for (int k = i; k < 3 * NBINS; k += stride) ws[k] = 0.0f;
    if (i < 3) ws[SCAL_OFF + i] = 0.0f;
}

// ---------------------------------------------------------------------------
// Kernel 1: streaming pass. B128 loads, LDS-privatized histograms (ds_add_f32),
// one global atomic flush per block per bin.
//   ws[0..2047]    s[t]  = sum exp(hr)
//   ws[2048..4095] T[t]  = sum_{events} exp(hr)
//   ws[4096..6143] n[t]  = event count
//   ws[8192] += sum hr*e ; ws[8193] += sum e
// ---------------------------------------------------------------------------
__global__ __launch_bounds__(256) void cox_accum(
    const float4* __restrict__ hr4, const int4* __restrict__ du4,
    const int4* __restrict__ ev4, float* __restrict__ ws, int nvec)
{
    __shared__ float ls[NBINS];
    __shared__ float lT[NBINS];
    __shared__ float ln[NBINS];
    __shared__ float red[16];

    for (int b = threadIdx.x; b < NBINS; b += blockDim.x) {
        ls[b] = 0.0f; lT[b] = 0.0f; ln[b] = 0.0f;
    }
    __syncthreads();

    float hrE = 0.0f, ecnt = 0.0f;
    const int stride = blockDim.x * gridDim.x;
    for (int idx = blockIdx.x * blockDim.x + threadIdx.x; idx < nvec; idx += stride) {
        // speculative prefetch of next tile (gfx1250 global_prefetch_b8)
        __builtin_prefetch(&hr4[idx + stride], 0, 0);
        __builtin_prefetch(&du4[idx + stride], 0, 0);
        __builtin_prefetch(&ev4[idx + stride], 0, 0);

        const float4 h = hr4[idx];
        const int4   d = du4[idx];
        const int4   e = ev4[idx];

        float ex;
        ex = __expf(h.x); atomicAdd(&ls[d.x], ex);
        if (e.x) { atomicAdd(&lT[d.x], ex); atomicAdd(&ln[d.x], 1.0f); hrE += h.x; ecnt += 1.0f; }
        ex = __expf(h.y); atomicAdd(&ls[d.y], ex);
        if (e.y) { atomicAdd(&lT[d.y], ex); atomicAdd(&ln[d.y], 1.0f); hrE += h.y; ecnt += 1.0f; }
        ex = __expf(h.z); atomicAdd(&ls[d.z], ex);
        if (e.z) { atomicAdd(&lT[d.z], ex); atomicAdd(&ln[d.z], 1.0f); hrE += h.z; ecnt += 1.0f; }
        ex = __expf(h.w); atomicAdd(&ls[d.w], ex);
        if (e.w) { atomicAdd(&lT[d.w], ex); atomicAdd(&ln[d.w], 1.0f); hrE += h.w; ecnt += 1.0f; }
    }
    __syncthreads();

    float* s_g = ws;
    float* T_g = ws + NBINS;
    float* n_g = ws + 2 * NBINS;
    for (int b = threadIdx.x; b < NBINS; b += blockDim.x) {
        float v0 = ls[b], v1 = lT[b], v2 = ln[b];
        if (v0 != 0.0f) unsafeAtomicAdd(&s_g[b], v0);
        if (v1 != 0.0f) unsafeAtomicAdd(&T_g[b], v1);
        if (v2 != 0.0f) unsafeAtomicAdd(&n_g[b], v2);
    }

    // wave32 shuffle reduction, then cross-wave via LDS
    for (int o = 16; o > 0; o >>= 1) {
        hrE  += __shfl_down(hrE,  o, 32);
        ecnt += __shfl_down(ecnt, o, 32);
    }
    const int wid = threadIdx.x >> 5, lane = threadIdx.x & 31;
    if (lane == 0) { red[wid] = hrE; red[8 + wid] = ecnt; }
    __syncthreads();
    if (threadIdx.x == 0) {
        float a = 0.0f, c = 0.0f;
        for (int w = 0; w < 8; ++w) { a += red[w]; c += red[8 + w]; }
        unsafeAtomicAdd(&ws[SCAL_OFF + 0], a);
        unsafeAtomicAdd(&ws[SCAL_OFF + 1], c);
    }
}

// ---------------------------------------------------------------------------
// Kernel 2: reverse cumsum R[t] = sum_{u>=t} s[u] via WMMA.
//   Reshape s -> A (128 rows x 16 cols).  W = A * U, U[k][j] = (k>=j) (ones
//   lower-triangular) gives intra-row suffix sums exactly (f32 adds, reordered
//   only).  Then a 128-row suffix pass stitches rows together.
//   One wave per 16-row tile; v_wmma_f32_16x16x4_f32 chained over K (4 x K=4).
// ---------------------------------------------------------------------------
__global__ __launch_bounds__(256) void cox_scan_wmma(float* __restrict__ ws)
{
    __shared__ float A[NBINS];
    __shared__ float W[NBINS];
    __shared__ float Suf[128];

    const float* s_g = ws;
    float* R_g = ws + 3 * NBINS;

    for (int t = threadIdx.x; t < NBINS; t += 256) A[t] = s_g[t];
    __syncthreads();

    const int wv   = threadIdx.x >> 5;   // wave id 0..7 -> 16-row tile
    const int lane = threadIdx.x & 31;
    const int half = lane >> 4;          // 0: lanes 0-15, 1: lanes 16-31
    const int m    = lane & 15;          // A: row-in-tile / B,C,D: column N
    const int rowBase = wv << 4;

    v8f acc = {};
#pragma unroll
    for (int c = 0; c < 4; ++c) {
        // A fragment (16x4 f32): lanes 0-15 hold K = 4c+0,4c+1; lanes 16-31 hold 4c+2,4c+3
        const int k0 = 4 * c + 2 * half;
        v2f a, b;
        a[0] = A[((rowBase + m) << 4) + k0];
        a[1] = A[((rowBase + m) << 4) + k0 + 1];
        // B fragment (4x16): U[k][j] = (k >= j), j = m
        b[0] = (k0     >= m) ? 1.0f : 0.0f;
        b[1] = (k0 + 1 >= m) ? 1.0f : 0.0f;
        acc = __builtin_amdgcn_wmma_f32_16x16x4_f32(
            /*neg_a=*/false, a, /*neg_b=*/false, b,
            /*c_mod=*/(short)0, acc, /*reuse_a=*/false, /*reuse_b=*/false);
    }

    // D layout: VGPR r -> row (r + 8*half) of the tile, col = m
#pragma unroll
    for (int r = 0; r < 8; ++r) {
        const int i = rowBase + r + (half << 3);
        W[(i << 4) + m] = acc[r];
    }
    __syncthreads();

    // Suf[i] = sum of full-row sums of rows > i.  Row sum == W[i][0].
    if (threadIdx.x == 0) {
        float run = 0.0f;
        for (int i = 127; i >= 0; --i) { Suf[i] = run; run += W[i << 4]; }
    }
    __syncthreads();

    for (int t = threadIdx.x; t < NBINS; t += 256)
        R_g[t] = W[t] + Suf[t >> 4];
}

// ---------------------------------------------------------------------------
// Kernel 3: Efron correction, one block per time t:
//   corr += sum_{k=0}^{n_t-1} log( R[t] - (k/n_t) * T[t] )
// ---------------------------------------------------------------------------
__global__ __launch_bounds__(256) void cox_corr(float* __restrict__ ws)
{
    const float* T_g = ws + NBINS;
    const float* n_g = ws + 2 * NBINS;
    const float* R_g = ws + 3 * NBINS;
    __shared__ float red[8];

    const int t = blockIdx.x;
    const float nf = n_g[t];
    const int ni = (int)(nf + 0.5f);

    float acc = 0.0f;
    if (ni > 0) {
        const float Rt = R_g[t], Tt = T_g[t], inv = 1.0f / nf;
        for (int k = threadIdx.x; k < ni; k += blockDim.x)
            acc += __logf(Rt - ((float)k * inv) * Tt);
    }
    for (int o = 16; o > 0; o >>= 1) acc += __shfl_down(acc, o, 32);
    const int wid = threadIdx.x >> 5, lane = threadIdx.x & 31;
    if (lane == 0) red[wid] = acc;
    __syncthreads();
    if (threadIdx.x == 0) {
        float s = 0.0f;
        for (int w = 0; w < 8; ++w) s += red[w];
        if (s != 0.0f) unsafeAtomicAdd(&ws[SCAL_OFF + 2], s);
    }
}

// ---------------------------------------------------------------------------
// Kernel 4: finalize scalar loss.
// ---------------------------------------------------------------------------
__global__ void cox_final(const float* __restrict__ ws, float* __restrict__ out)
{
    const float hrE  = ws[SCAL_OFF + 0];
    const float eCnt = ws[SCAL_OFF + 1];
    const float corr = ws[SCAL_OFF + 2];
    out[0] = -(hrE - corr) / (eCnt + 1e-7f);
}

// ---------------------------------------------------------------------------
extern "C" void kernel_launch(void* const* d_in, const int* in_sizes, int n_in,
                              void* d_out, int out_size, void* d_ws, size_t ws_size,
                              hipStream_t stream)
{
    const float4* hr4 = (const float4*)d_in[0];  // hazard_ratio [N,1] f32
    const int4*   du4 = (const int4*)  d_in[1];  // durations    [N]   i32
    const int4*   ev4 = (const int4*)  d_in[2];  // events       [N]   i32
    float* ws  = (float*)d_ws;
    float* out = (float*)d_out;

    const int n    = in_sizes[1];
    const int nvec = n >> 2;                     // N divisible by 4

    cox_init <<<8,    256, 0, stream>>>(ws);
    cox_accum<<<512,  256, 0, stream>>>(hr4, du4, ev4, ws, nvec);
    cox_scan_wmma<<<1, 256, 0, stream>>>(ws);
    cox_corr <<<NBINS, 256, 0, stream>>>(ws);
    cox_final<<<1, 1, 0, stream>>>(ws, out);
}